// SDE_15479062134932
// MI455X (gfx1250) — compile-verified
//
#include <hip/hip_runtime.h>
#include <hip/hip_bf16.h>
#include <math.h>

// ---------------------------------------------------------------------------
// SDE scan (Izhikevich Euler-Maruyama), batch=2048 chains, ~20000 steps.
// Latency-bound streaming: only 64 wave32 exist, so throughput = bytes in
// flight / HBM latency. Strategy: single-wave workgroups (barriers -> S_NOP),
// 8-deep async global->LDS pipeline per wave (56 outstanding b128 DMA ops,
// just under the 6-bit ASYNCcnt cap), NT b64 stores for the 328MB output.
// No matmul shape -> WMMA inapplicable; CDNA5 paths used: ASYNCcnt DMA,
// s_wait_asynccnt / s_wait_dscnt split counters, ds_load, NT cache hints.
// ---------------------------------------------------------------------------

#define BLOCK 32           // one wave32 per workgroup: private LDS + ASYNCcnt
#define TILE  32           // time steps per LDS tile (TILE x 32 floats = 4KB)
#define NBUF  8            // pipeline depth: 7 tiles (56 ops, 28KB) in flight
#define OPS_PER_TILE (TILE / 4)   // per-thread async b128 ops per tile = 8

typedef __attribute__((ext_vector_type(2))) float f32x2;
typedef __attribute__((ext_vector_type(4))) float f32x4;
typedef __attribute__((ext_vector_type(4))) int   i32x4;

#if defined(__gfx1250__) && __has_builtin(__builtin_amdgcn_global_load_async_to_lds_b128) && __has_builtin(__builtin_amdgcn_s_wait_asynccnt)
#define USE_ASYNC 1
#else
#define USE_ASYNC 0
#endif

#if USE_ASYNC
typedef i32x4 __attribute__((address_space(1)))* as1_v4i;   // global int4*
typedef i32x4 __attribute__((address_space(3)))* as3_v4i;   // LDS    int4*
#endif

__device__ __forceinline__ void g2lds_b128(const float* __restrict__ g, float* lds_dst) {
#if USE_ASYNC
    __builtin_amdgcn_global_load_async_to_lds_b128(
        (as1_v4i)(const void*)g, (as3_v4i)(void*)lds_dst, 0, 0);
#else
    *(f32x4*)lds_dst = *(const f32x4*)g;
#endif
}

template <int N>
__device__ __forceinline__ void wait_async() {
#if USE_ASYNC
    __builtin_amdgcn_s_wait_asynccnt(N);
#endif
}

__device__ __forceinline__ void wait_ds0() {
#if USE_ASYNC
#if __has_builtin(__builtin_amdgcn_s_wait_dscnt)
    __builtin_amdgcn_s_wait_dscnt(0);
#else
    asm volatile("s_wait_dscnt 0" ::: "memory");
#endif
#endif
}

// Descending immediate ladder for the pipeline drain (imm-only s_wait).
__device__ __forceinline__ void wait_remaining(int m) {
    if      (m >= 6) wait_async<6 * OPS_PER_TILE>();
    else if (m == 5) wait_async<5 * OPS_PER_TILE>();
    else if (m == 4) wait_async<4 * OPS_PER_TILE>();
    else if (m == 3) wait_async<3 * OPS_PER_TILE>();
    else if (m == 2) wait_async<2 * OPS_PER_TILE>();
    else if (m == 1) wait_async<1 * OPS_PER_TILE>();
    else             wait_async<0>();
}

// ---------------------------------------------------------------------------
// Pre-pass: ext[t] = 10*(tanh(theta[idx(t)])+1); float64 index math replicates
// the reference's segment search + floor indexing bit-exactly.
// ---------------------------------------------------------------------------
__global__ void sde_ext_kernel(const float* __restrict__ theta,
                               float* __restrict__ ext,
                               int num_steps, int theta_len) {
    int t = blockIdx.x * blockDim.x + threadIdx.x;
    if (t >= num_steps) return;

    const double Hd = 20.0 / 20000.0;
    double ts = (double)t * Hd;

    int seg = 0;
    double sep_prev = 0.0;
#pragma unroll
    for (int i = 0; i < 9; ++i) {                // SEPARATE = 2,4,...,18
        double s = 2.0 * (double)(i + 1);
        if (ts >= s) { seg = i + 1; sep_prev = s; }
    }
    long long start = (long long)seg * 2000;     // ceil(sep/H) == 2000*seg
    long long local = (long long)floor((ts - sep_prev) / 20.0 * 20001.0);
    long long idx = start + local;
    if (idx < 0) idx = 0;
    if (idx > 20000) idx = 20000;                // clip to T_SIZE-1
    if (idx >= theta_len) idx = theta_len - 1;

    float th = theta[idx];
    ext[t] = 10.0f * (tanhf(th) + 1.0f);
}

// ---------------------------------------------------------------------------
// Per-step state update (fp32, matches jnp.float32 reference math).
// ---------------------------------------------------------------------------
__device__ __forceinline__ void sde_step(float& yv, float& yu, float w, float e) {
    const float Hf    = 0.001f;
    const float sqrtH = 0.031622776601683794f;   // float(sqrt(0.001))

    float vv = yv + 65.0f;                       // yv - C, C = -65
    float q  = floorf(vv / 95.0f);               // TH - C = 95
    float v  = vv - 95.0f * q - 65.0f;           // wrapped v
    float u  = yu + 8.0f * q;                    // u + D*num_fire

    float vp = 0.04f * v * v + 5.0f * v + 140.0f - u + 5.0f + e;
    vp = fmaxf(vp, 0.0f);
    float up = 0.02f * (0.2f * v - u);           // a*(b*v - u)

    yv = yv + Hf * vp + sqrtH * w;
    yu = yu + Hf * up;
}

// ---------------------------------------------------------------------------
// Main scan: one wave32 per block, one chain per lane. dW staged via 8-deep
// double-buffer ring of async b128 DMA; no barriers (single-wave workgroup).
// Requires batch % 32 == 0 (reference: 2048 -> 64 blocks).
// ---------------------------------------------------------------------------
__global__ __launch_bounds__(BLOCK)
void sde_scan_kernel(const float* __restrict__ y0,
                     const float* __restrict__ dW,
                     const float* __restrict__ ext,
                     float* __restrict__ out,
                     int batch, int num_steps) {
    __shared__ float sdw[NBUF][TILE * BLOCK];    // 32 KB of 320 KB WGP LDS

    const int tid = threadIdx.x;
    const int b0  = blockIdx.x * BLOCK;
    const int gb  = b0 + tid;

    float yv = y0[2 * gb + 0];
    float yu = y0[2 * gb + 1];

    f32x2* out2 = (f32x2*)out;
    {
        f32x2 o; o.x = yv; o.y = yu;             // row 0 of the output is y0
        __builtin_nontemporal_store(o, &out2[gb]);
    }

    const int ntiles = (num_steps + TILE - 1) / TILE;

    // Issue one tile's async copies: TILE rows x 32 floats (128B/row).
    auto issue_tile = [&](int k) {
        const int t0 = k * TILE;
        float* dst = &sdw[k & (NBUF - 1)][0];
#pragma unroll
        for (int j = 0; j < OPS_PER_TILE; ++j) {
            int i4   = j * BLOCK + tid;          // float4 index within tile
            int row  = i4 >> 3;                  // BLOCK/4 = 8 float4 per row
            int col4 = i4 & 7;
            int t = t0 + row;
            if (t >= num_steps) t = t0;          // pad rows: harmless reload
            const float* gsrc = dW + (size_t)t * (size_t)batch + b0 + col4 * 4;
            g2lds_b128(gsrc, dst + i4 * 4);
        }
    };

    // Prologue: fill pipeline with up to NBUF-1 tiles (56 async ops <= 63 cap).
    for (int k = 0; k < NBUF - 1 && k < ntiles; ++k) issue_tile(k);

    const int kmain = (ntiles > (NBUF - 1)) ? (ntiles - (NBUF - 1)) : 0;

    // ---- steady state: full tiles, constant trip count, no guards ----------
    for (int k = 0; k < kmain; ++k) {
        wait_async<(NBUF - 2) * OPS_PER_TILE>(); // tile k landed in LDS
        wait_ds0();                              // WAR: ring slot reads done
        issue_tile(k + NBUF - 1);                // refill pipeline (<=56 out)

        const float* buf = &sdw[k & (NBUF - 1)][0];
        const int t0 = k * TILE;
#pragma unroll 8
        for (int r = 0; r < TILE; ++r) {
            const int t = t0 + r;
            float w = buf[r * BLOCK + tid];      // ds_load_b32, conflict-free
            float e = ext[t];                    // lane-uniform scalar load
            sde_step(yv, yu, w, e);
            f32x2 o; o.x = yv; o.y = yu;
            __builtin_nontemporal_store(o, &out2[(size_t)(t + 1) * (size_t)batch + gb]);
        }
    }

    // ---- drain: last <= NBUF-1 tiles, guarded, descending waits ------------
    for (int k = kmain; k < ntiles; ++k) {
        wait_remaining(ntiles - 1 - k);          // ensure tile k complete

        const float* buf = &sdw[k & (NBUF - 1)][0];
        const int t0 = k * TILE;
        int tmax = num_steps - t0;
        if (tmax > TILE) tmax = TILE;
        for (int r = 0; r < tmax; ++r) {
            const int t = t0 + r;
            float w = buf[r * BLOCK + tid];
            float e = ext[t];
            sde_step(yv, yu, w, e);
            f32x2 o; o.x = yv; o.y = yu;
            __builtin_nontemporal_store(o, &out2[(size_t)(t + 1) * (size_t)batch + gb]);
        }
    }
}

// ---------------------------------------------------------------------------
extern "C" void kernel_launch(void* const* d_in, const int* in_sizes, int n_in,
                              void* d_out, int out_size, void* d_ws, size_t ws_size,
                              hipStream_t stream) {
    const float* y0    = (const float*)d_in[0];  // [batch, 2]
    const float* theta = (const float*)d_in[1];  // [T_SIZE, 1]
    const float* dW    = (const float*)d_in[2];  // [num_steps, batch]
    float* out = (float*)d_out;                  // [num_steps+1, batch, 2]

    const int batch     = in_sizes[0] / 2;       // 2048
    const int num_steps = in_sizes[2] / batch;   // 20000
    const int theta_len = in_sizes[1];           // 20001

    float* ext = (float*)d_ws;                   // num_steps floats of scratch

    sde_ext_kernel<<<(num_steps + 255) / 256, 256, 0, stream>>>(
        theta, ext, num_steps, theta_len);

    const int nblocks = batch / BLOCK;           // 64 single-wave blocks
    sde_scan_kernel<<<nblocks, BLOCK, 0, stream>>>(
        y0, dW, ext, out, batch, num_steps);
}